// UniCDR_22196390986325
// MI455X (gfx1250) — compile-verified
//
#include <hip/hip_runtime.h>
#include <math.h>

#define N_USERS 100000
#define N_ITEMS 50001
#define N_EDGES 1000000
#define DIM     64
#define BATCH   4096
#define MAXLEN  50
#define LAMBDA_A 0.5f

typedef __attribute__((ext_vector_type(2))) float v2f;
typedef __attribute__((ext_vector_type(8))) float v8f;
typedef int v4i __attribute__((vector_size(4 * sizeof(int))));

// Direct global->LDS async copies (gfx1250 GLOBAL_LOAD_ASYNC_TO_LDS_B128,
// ASYNCcnt-tracked). Guarded so the file compiles on toolchains that do not
// declare the builtins (falls back to VGPR-staged float4 copies).
#if defined(__AMDGCN__) && \
    __has_builtin(__builtin_amdgcn_global_load_async_to_lds_b128) && \
    __has_builtin(__builtin_amdgcn_s_wait_asynccnt)
#define USE_ASYNC_LDS 1
#else
#define USE_ASYNC_LDS 0
#endif

#if USE_ASYNC_LDS
#define GLOBAL_AS __attribute__((address_space(1)))
#define LOCAL_AS  __attribute__((address_space(3)))
__device__ __forceinline__ void async_copy16(const float* g, float* l)
{
    // Builtin signature: (global int32x4*, local int32x4*, imm offset, imm cpol).
    // int->pointer casts avoid generic<->AS addrspacecast diagnostics;
    // generic LDS pointers carry the byte offset in their low 32 bits.
    __builtin_amdgcn_global_load_async_to_lds_b128(
        (GLOBAL_AS v4i*)(uintptr_t)g,
        (LOCAL_AS  v4i*)(uintptr_t)(uint32_t)(uintptr_t)l,
        /*offset=*/0, /*cpol=*/0);
}
#endif

// ---------------------------------------------------------------------------
// SpMM edge pass: dst[dst_idx[e]] += w[e] * src[src_idx[e]], rows of DIM=64.
// 32 lanes per edge, 2 floats per lane (float2 load, two f32 global atomics).
// Row tables are L2-resident (192 MB L2 >> 25.6 MB + 12.8 MB tables).
// ---------------------------------------------------------------------------
__global__ __launch_bounds__(256) void spmm_edge_kernel(
    const float* __restrict__ src_rows,
    const float* __restrict__ edge_w,
    const int* __restrict__ src_idx,
    const int* __restrict__ dst_idx,
    float* __restrict__ dst_rows)
{
    int t = blockIdx.x * blockDim.x + threadIdx.x;
    int edge = t >> 5;
    if (edge >= N_EDGES) return;
    int d = (t & 31) << 1;

    float w = edge_w[edge];
    int s = src_idx[edge];
    int dd = dst_idx[edge];

    const float2* vp = (const float2*)(src_rows + (size_t)s * DIM + d);
    float2 v = *vp;
    float* outp = dst_rows + (size_t)dd * DIM + d;
    atomicAdd(outp + 0, w * v.x);
    atomicAdd(outp + 1, w * v.y);
}

// ---------------------------------------------------------------------------
// V_final = mean(V0, V1, V2) + V0
// ---------------------------------------------------------------------------
__global__ __launch_bounds__(256) void combine_layers_kernel(
    const float* __restrict__ V0, const float* __restrict__ V1,
    const float* __restrict__ V2, float* __restrict__ Vfin)
{
    int i = blockIdx.x * blockDim.x + threadIdx.x;
    if (i >= N_ITEMS * DIM) return;
    float v0 = V0[i];
    Vfin[i] = v0 + (v0 + V1[i] + V2[i]) * (1.0f / 3.0f);
}

// ---------------------------------------------------------------------------
// Fused attention pooling, one block (4 waves, wave32) per batch row.
//   key  = tanh(seq[64x64] @ W_attT[64x64] + b)   via V_WMMA_F32_16X16X4_F32
//   att  = key . id_emb, masked softmax over 50 positions
//   pooled = attn @ seq ;  out = pooled @ W_aggT ;  res = 0.5*id + 0.5*out
//
// WMMA fragment layouts (ISA 7.12.2, wave32):
//   A (16x4 f32, 2 VGPRs): lane holds (m = lane%16, k = 2*(lane/16) + {0,1})
//   B (4x16 f32, 2 VGPRs): lane holds (k = 2*(lane/16) + {0,1}, n = lane%16)
//     B[k][n] = W_att_w[n][k] -> read row-major sW as sW[n*PITCH + k]
//   C (16x16 f32, 8 VGPRs): vgpr j -> row m0 + j + 8*(lane/16), col lane%16
// ---------------------------------------------------------------------------
#define LDS_PITCH 68  // rows 272 B: 16 B-aligned for B128; bank = (4r+c)%64

__global__ __launch_bounds__(128) void attention_pool_kernel(
    const float* __restrict__ Vfin,
    const float* __restrict__ user_table,
    const float* __restrict__ W_att_w,
    const float* __restrict__ W_att_b,
    const float* __restrict__ W_agg,
    const int* __restrict__ user_idx,
    const int* __restrict__ ctx_items,
    float* __restrict__ out)
{
    __shared__ __align__(16) float sSeq[64 * LDS_PITCH];
    __shared__ __align__(16) float sW[64 * LDS_PITCH];   // row-major W_att_w
    __shared__ __align__(16) float sKey[64 * LDS_PITCH];
    __shared__ __align__(16) float sId[64];
    __shared__ float sAttn[64];
    __shared__ float sPooled[64];
    __shared__ int   sRow[64];
    __shared__ float sDen;

    const int b = blockIdx.x;
    const int tid = threadIdx.x;       // 0..127
    const int wave = tid >> 5;         // 0..3 -> N stripe
    const int lane = tid & 31;

    // ---- stage context item indices ----
    if (tid < 64) sRow[tid] = (tid < MAXLEN) ? ctx_items[b * MAXLEN + tid] : 0;
    __syncthreads();

    const int uidx = user_idx[b];

    // ---- stage W (row-major), id_emb, and gathered seq rows into LDS ----
    // 64 rows x 16 float4 chunks each for W and seq; 16 chunks for id_emb.
#if USE_ASYNC_LDS
    for (int i = tid; i < 64 * 16; i += 128) {
        int n = i >> 4, c = (i & 15) << 2;
        async_copy16(W_att_w + n * DIM + c, &sW[n * LDS_PITCH + c]);
    }
    if (tid < 16)
        async_copy16(user_table + (size_t)uidx * DIM + (tid << 2), &sId[tid << 2]);
    for (int i = tid; i < 64 * 16; i += 128) {
        int l = i >> 4, c = (i & 15) << 2;
        if (l < MAXLEN) {
            async_copy16(Vfin + (size_t)sRow[l] * DIM + c, &sSeq[l * LDS_PITCH + c]);
        } else {
            float4 z = make_float4(0.f, 0.f, 0.f, 0.f);
            *(float4*)&sSeq[l * LDS_PITCH + c] = z;     // zero padding rows
        }
    }
    __builtin_amdgcn_s_wait_asynccnt(0);
#else
    for (int i = tid; i < 64 * 16; i += 128) {
        int n = i >> 4, c = (i & 15) << 2;
        *(float4*)&sW[n * LDS_PITCH + c] = *(const float4*)(W_att_w + n * DIM + c);
    }
    if (tid < 16)
        *(float4*)&sId[tid << 2] =
            *(const float4*)(user_table + (size_t)uidx * DIM + (tid << 2));
    for (int i = tid; i < 64 * 16; i += 128) {
        int l = i >> 4, c = (i & 15) << 2;
        float4 v = make_float4(0.f, 0.f, 0.f, 0.f);
        if (l < MAXLEN) v = *(const float4*)(Vfin + (size_t)sRow[l] * DIM + c);
        *(float4*)&sSeq[l * LDS_PITCH + c] = v;
    }
#endif
    __syncthreads();

    // ---- key = tanh(seq @ W^T + bias) via fp32 WMMA ----
    const int n0 = wave * 16;
    const int mrow = lane & 15;
    const int khalf = (lane >> 4) << 1;     // 0 or 2
    const float bias = W_att_b[n0 + mrow];

    for (int mt = 0; mt < 4; ++mt) {
        const int m0 = mt * 16;
        v8f acc = {};
#pragma unroll
        for (int k0 = 0; k0 < 64; k0 += 4) {
            v2f a, bb;
            a.x  = sSeq[(m0 + mrow) * LDS_PITCH + k0 + khalf];
            a.y  = sSeq[(m0 + mrow) * LDS_PITCH + k0 + khalf + 1];
            bb.x = sW[(n0 + mrow) * LDS_PITCH + k0 + khalf];      // B[k][n]=W[n][k]
            bb.y = sW[(n0 + mrow) * LDS_PITCH + k0 + khalf + 1];
            acc = __builtin_amdgcn_wmma_f32_16x16x4_f32(
                /*neg_a=*/false, a, /*neg_b=*/false, bb,
                /*c_mod=*/(short)0, acc, /*reuse_a=*/false, /*reuse_b=*/false);
        }
        const int mbase = m0 + ((lane >> 4) << 3);
#pragma unroll
        for (int j = 0; j < 8; ++j)
            sKey[(mbase + j) * LDS_PITCH + n0 + mrow] = tanhf(acc[j] + bias);
    }
    __syncthreads();

    // ---- masked attention logits: att[l] = key[l] . id_emb ; mask rowsum==0
    if (tid < MAXLEN) {
        float dot = 0.0f, rs = 0.0f;
#pragma unroll 8
        for (int d = 0; d < DIM; ++d) {
            dot += sKey[tid * LDS_PITCH + d] * sId[d];
            rs  += sSeq[tid * LDS_PITCH + d];
        }
        float x = (rs == 0.0f) ? 0.0f : dot;
        sAttn[tid] = __expf(x);
    }
    __syncthreads();

    if (tid == 0) {
        float s = 0.0f;
        for (int l = 0; l < MAXLEN; ++l) s += sAttn[l];
        sDen = s + 1e-12f;
    }
    __syncthreads();

    // ---- pooled[d] = sum_l attn[l] * seq[l][d] ----
    if (tid < 64) {
        float inv = 1.0f / sDen;
        float p = 0.0f;
#pragma unroll 5
        for (int l = 0; l < MAXLEN; ++l)
            p += sAttn[l] * inv * sSeq[l * LDS_PITCH + tid];
        sPooled[tid] = p;
    }
    __syncthreads();

    // ---- out[e] = pooled . W_agg[e][:] ; blend with id_emb ----
    if (tid < 64) {
        float o = 0.0f;
#pragma unroll 8
        for (int d = 0; d < DIM; ++d)
            o += sPooled[d] * W_agg[tid * DIM + d];
        out[(size_t)b * DIM + tid] = LAMBDA_A * sId[tid] + (1.0f - LAMBDA_A) * o;
    }
}

// ---------------------------------------------------------------------------
extern "C" void kernel_launch(void* const* d_in, const int* in_sizes, int n_in,
                              void* d_out, int out_size, void* d_ws, size_t ws_size,
                              hipStream_t stream)
{
    const float* user_table = (const float*)d_in[0];
    const float* item_table = (const float*)d_in[1];
    const float* ev_uv      = (const float*)d_in[2];
    const float* ev_vu      = (const float*)d_in[3];
    const float* W_att_w    = (const float*)d_in[4];
    const float* W_att_b    = (const float*)d_in[5];
    const float* W_agg      = (const float*)d_in[6];
    const int*   edge_u     = (const int*)d_in[7];
    const int*   edge_i     = (const int*)d_in[8];
    const int*   user_idx   = (const int*)d_in[9];
    const int*   ctx_items  = (const int*)d_in[10];
    float* out = (float*)d_out;

    // workspace carve-up (~64 MB total)
    float* umsg = (float*)d_ws;                       // [N_USERS  x 64]
    float* V1   = umsg + (size_t)N_USERS * DIM;       // [N_ITEMS x 64]
    float* V2   = V1   + (size_t)N_ITEMS * DIM;       // [N_ITEMS x 64]
    float* Vfin = V2   + (size_t)N_ITEMS * DIM;       // [N_ITEMS x 64]

    const size_t umsg_bytes = sizeof(float) * (size_t)N_USERS * DIM;
    const size_t item_bytes = sizeof(float) * (size_t)N_ITEMS * DIM;

    const int spmm_threads = N_EDGES * 32;
    const dim3 spmm_grid((spmm_threads + 255) / 256);
    const dim3 comb_grid((N_ITEMS * DIM + 255) / 256);

    // ---- GNN layer 1: V1 = VU @ (UV @ V0) ----
    (void)hipMemsetAsync(umsg, 0, umsg_bytes, stream);
    spmm_edge_kernel<<<spmm_grid, 256, 0, stream>>>(item_table, ev_uv, edge_i, edge_u, umsg);
    (void)hipMemsetAsync(V1, 0, item_bytes, stream);
    spmm_edge_kernel<<<spmm_grid, 256, 0, stream>>>(umsg, ev_vu, edge_u, edge_i, V1);

    // ---- GNN layer 2: V2 = VU @ (UV @ V1) ----
    (void)hipMemsetAsync(umsg, 0, umsg_bytes, stream);
    spmm_edge_kernel<<<spmm_grid, 256, 0, stream>>>(V1, ev_uv, edge_i, edge_u, umsg);
    (void)hipMemsetAsync(V2, 0, item_bytes, stream);
    spmm_edge_kernel<<<spmm_grid, 256, 0, stream>>>(umsg, ev_vu, edge_u, edge_i, V2);

    // ---- Vfin = mean(V0,V1,V2) + V0 ----
    combine_layers_kernel<<<comb_grid, 256, 0, stream>>>(item_table, V1, V2, Vfin);

    // ---- fused attention pooling (WMMA + async-LDS staging) ----
    attention_pool_kernel<<<BATCH, 128, 0, stream>>>(
        Vfin, user_table, W_att_w, W_att_b, W_agg, user_idx, ctx_items, out);
}